// ScaledDotProductAttention_69715909149279
// MI455X (gfx1250) — compile-verified
//
#include <hip/hip_runtime.h>
#include <hip/hip_bf16.h>

typedef __attribute__((ext_vector_type(2))) float v2f;
typedef __attribute__((ext_vector_type(4))) float v4f;
typedef __attribute__((ext_vector_type(8))) float v8f;

#define QLEN 1024
#define SLEN 1024
#define DK   64
#define SST  1028   // padded LDS row stride (floats) for the 16 x 1024 score buffer

__global__ __launch_bounds__(256)
void ScaledDotProductAttention_69715909149279_kernel(
    const float* __restrict__ Q, const float* __restrict__ K,
    const float* __restrict__ V, const float* __restrict__ PREV,
    const float* __restrict__ scale_p,
    float* __restrict__ outO, float* __restrict__ outW, float* __restrict__ outS)
{
    extern __shared__ float smem[];
    float* sQ     = smem;               // 16*64   = 1024 floats
    float* sS     = sQ + 16 * DK;       // 16*1028 = 16448 floats
    float* sO     = sS + 16 * SST;      // 16*64   = 1024 floats
    float* red    = sO + 16 * DK;       // 256 floats
    float* rowmax = red + 256;          // 16 floats
    float* rowinv = rowmax + 16;        // 16 floats

    const int tid  = threadIdx.x;
    const int wave = tid >> 5;
    const int lane = tid & 31;
    const int l16  = lane & 15;
    const int h16  = lane >> 4;         // which half of the wave (K/M split in WMMA layouts)

    const int bh = blockIdx.x >> 6;     // 64 query tiles per (b,h)
    const int qt = blockIdx.x & 63;
    const int q0 = qt * 16;

    const float scale = scale_p[0];

    const float* Qb = Q    + ((size_t)bh * QLEN + q0) * DK;
    const float* Kb = K    + (size_t)bh * DK * SLEN;
    const float* Vb = V    + (size_t)bh * SLEN * DK;
    const float* Pb = PREV + ((size_t)bh * QLEN + q0) * (size_t)SLEN;
    float* Ob = outO + ((size_t)bh * QLEN + q0) * DK;
    float* Wb = outW + ((size_t)bh * QLEN + q0) * (size_t)SLEN;
    float* Sb = outS + ((size_t)bh * QLEN + q0) * (size_t)SLEN;

    // ---- load Q tile to LDS, zero the output accumulator tile ----
    {
        const v4f* qs = (const v4f*)Qb;
        ((v4f*)sQ)[tid] = qs[tid];                    // 1024 floats = 256 float4
        ((v4f*)sO)[tid] = (v4f){0.f, 0.f, 0.f, 0.f};
    }
    __syncthreads();

    // ---- Phase 1: S = Q*K * scale + prev ; each wave owns 8 N-tiles of 16 ----
    for (int t = 0; t < 8; ++t) {
        const int n0 = (wave * 8 + t) * 16;
        v8f acc = {0.f, 0.f, 0.f, 0.f, 0.f, 0.f, 0.f, 0.f};
        const float* sQrow = sQ + l16 * DK + 2 * h16;          // A frag: row M=l16, K pair
        const float* Kcol  = Kb + (2 * h16) * SLEN + n0 + l16; // B frag: K row pair, col N=l16
        #pragma unroll
        for (int kb = 0; kb < DK / 4; ++kb) {
            v2f a = *(const v2f*)(sQrow + kb * 4);
            v2f b;
            b.x = Kcol[(kb * 4    ) * SLEN];
            b.y = Kcol[(kb * 4 + 1) * SLEN];
            acc = __builtin_amdgcn_wmma_f32_16x16x4_f32(
                false, a, false, b, (short)0, acc, false, false);
        }
        #pragma unroll
        for (int r = 0; r < 8; ++r) {
            const int row = r + 8 * h16;                       // C/D layout: vgpr r, lane half
            const int col = n0 + l16;
            // prev is single-use streaming data: NT load keeps L2 for K/V
            const float pv = __builtin_nontemporal_load(&Pb[(size_t)row * SLEN + col]);
            const float sv = acc[r] * scale + pv;
            sS[row * SST + col] = sv;
            // scores are write-once streaming output: NT store
            __builtin_nontemporal_store(sv, &Sb[(size_t)row * SLEN + col]);
        }
    }
    __syncthreads();

    // ---- Phase 2: row-wise softmax over sS (16 x 1024), write weights ----
    {
        const int r = tid >> 4;
        const int s = tid & 15;
        float m = -3.402823466e+38f;
        for (int c = s; c < SLEN; c += 16)
            m = fmaxf(m, sS[r * SST + c]);
        red[tid] = m;
        __syncthreads();
        if (tid < 16) {
            float mm = red[tid * 16];
            #pragma unroll
            for (int i = 1; i < 16; ++i) mm = fmaxf(mm, red[tid * 16 + i]);
            rowmax[tid] = mm;
        }
        __syncthreads();
        const float rm = rowmax[r];
        float sum = 0.f;
        for (int c = s; c < SLEN; c += 16) {
            const float w = __expf(sS[r * SST + c] - rm);
            sS[r * SST + c] = w;
            sum += w;
        }
        red[tid] = sum;
        __syncthreads();
        if (tid < 16) {
            float ss = 0.f;
            #pragma unroll
            for (int i = 0; i < 16; ++i) ss += red[tid * 16 + i];
            rowinv[tid] = 1.0f / ss;
        }
        __syncthreads();
        // normalize in LDS + coalesced NT global store of weights
        for (int j = 0; j < (16 * SLEN) / 256; ++j) {
            const int idx = tid + 256 * j;
            const int rr = idx >> 10;
            const int cc = idx & (SLEN - 1);
            const float w = sS[rr * SST + cc] * rowinv[rr];
            sS[rr * SST + cc] = w;
            __builtin_nontemporal_store(w, &Wb[(size_t)rr * SLEN + cc]);
        }
    }
    __syncthreads();

    // ---- Phase 3: O = W @ V ; wave -> d-tile (wave&3), slen half (wave>>2) ----
    {
        const int n0 = (wave & 3) * 16;
        const int kh = wave >> 2;
        v8f acc = {0.f, 0.f, 0.f, 0.f, 0.f, 0.f, 0.f, 0.f};
        const float* sWrow = sS + l16 * SST + 2 * h16;
        const float* Vcol  = Vb + (2 * h16) * DK + n0 + l16;
        #pragma unroll 4
        for (int kb = 0; kb < 512 / 4; ++kb) {
            const int kbase = kh * 512 + kb * 4;
            v2f a = *(const v2f*)(sWrow + kbase);
            v2f b;
            b.x = Vcol[(kbase    ) * DK];
            b.y = Vcol[(kbase + 1) * DK];
            acc = __builtin_amdgcn_wmma_f32_16x16x4_f32(
                false, a, false, b, (short)0, acc, false, false);
        }
        #pragma unroll
        for (int r = 0; r < 8; ++r) {
            const int row = r + 8 * h16;
            atomicAdd(&sO[row * DK + n0 + l16], acc[r]);   // ds_add_f32, merges the 2 halves
        }
    }
    __syncthreads();

    // ---- store output tile (16 x 64, contiguous, single-use -> NT) ----
    {
        const v4f o = ((const v4f*)sO)[tid];
        __builtin_nontemporal_store(o, &((v4f*)Ob)[tid]);
    }
}

extern "C" void kernel_launch(void* const* d_in, const int* in_sizes, int n_in,
                              void* d_out, int out_size, void* d_ws, size_t ws_size,
                              hipStream_t stream) {
    (void)in_sizes; (void)n_in; (void)d_ws; (void)ws_size; (void)out_size;

    const float* q     = (const float*)d_in[0];
    const float* k     = (const float*)d_in[1];
    const float* v     = (const float*)d_in[2];
    const float* prev  = (const float*)d_in[3];
    const float* scale = (const float*)d_in[4];

    const size_t OSZ = (size_t)2 * 16 * 1024 * 64;       // output elems
    const size_t WSZ = (size_t)2 * 16 * 1024 * 1024;     // weights elems

    float* outO = (float*)d_out;
    float* outW = outO + OSZ;
    float* outS = outW + WSZ;

    const size_t shmem = (size_t)(16 * 64 + 16 * SST + 16 * 64 + 256 + 32) * sizeof(float);
    (void)hipFuncSetAttribute((const void*)ScaledDotProductAttention_69715909149279_kernel,
                              hipFuncAttributeMaxDynamicSharedMemorySize, (int)shmem);

    const int blocks = 2 * 16 * (1024 / 16);             // 2048
    ScaledDotProductAttention_69715909149279_kernel<<<blocks, 256, shmem, stream>>>(
        q, k, v, prev, scale, outO, outW, outS);
}